// slideLayer_48155173322753
// MI455X (gfx1250) — compile-verified
//
#include <hip/hip_runtime.h>

typedef __attribute__((ext_vector_type(2))) float v2f;
typedef __attribute__((ext_vector_type(8))) float v8f;

constexpr int kB     = 64;
constexpr int kDIn   = 1024;
constexpr int kNAIn  = 512;
constexpr int kNAOut = 4096;

constexpr int kWavesPerBlock  = 8;                       // 256 threads, wave32
constexpr int kJPerBlock      = kWavesPerBlock * 16;     // 128 outputs per block
constexpr int kBlocksPerBatch = kNAOut / kJPerBlock;     // 32

// One wave computes a 16-row output tile with V_WMMA_F32_16X16X4_F32.
// A (16x4)  : per-K-step gathered weight elems; lane<16 -> m=lane, k={0,1};
//             lane>=16 -> m=lane-16, k={2,3}  (ISA 32-bit A layout).
// B (4x16)  : x_g values replicated along N   (VGPR0=k_even, VGPR1=k_odd).
// D (16x16) : dot products replicated along N; extract N=0 (lanes 0 & 16).
__global__ __launch_bounds__(256) void slide_gather_gemv_wmma(
    const float* __restrict__ in_values,   // [64, 1024]
    const float* __restrict__ weight,      // [131072, 1024]
    const float* __restrict__ bias,        // [131072]
    const int*   __restrict__ a_in,        // [64, 512]
    const int*   __restrict__ a_out,       // [64, 4096]
    float*       __restrict__ out_values)  // [64, 4096]
{
    __shared__ int   s_idx[kNAIn];
    __shared__ float s_x[kNAIn];

    const int b    = blockIdx.x / kBlocksPerBatch;
    const int jblk = blockIdx.x % kBlocksPerBatch;
    const int t    = threadIdx.x;

    // Stage active-input indices and gathered x for this batch into LDS.
    for (int i = t; i < kNAIn; i += 256) {
        const int idx = a_in[b * kNAIn + i];
        s_idx[i] = idx;
        s_x[i]   = in_values[b * kDIn + idx];
    }
    __syncthreads();

    const int wave  = t >> 5;
    const int lane  = t & 31;
    const int jbase = jblk * kJPerBlock + wave * 16;

    const int m    = lane & 15;
    const int koff = (lane < 16) ? 0 : 2;

    const int    orow = a_out[b * kNAOut + jbase + m];
    const float* wrow = weight + (size_t)orow * kDIn;

    v8f acc = {};
#pragma unroll 4
    for (int kk = 0; kk < kNAIn; kk += 4) {
        const int i0 = s_idx[kk + koff];
        const int i1 = s_idx[kk + koff + 1];
        v2f a;
        a.x = wrow[i0];                 // A[m][koff]
        a.y = wrow[i1];                 // A[m][koff+1]
        v2f bx;
        bx.x = s_x[kk + koff];          // B[koff][*]   (replicated over N)
        bx.y = s_x[kk + koff + 1];      // B[koff+1][*]
        acc = __builtin_amdgcn_wmma_f32_16x16x4_f32(
            /*neg_a=*/false, a, /*neg_b=*/false, bx,
            /*c_mod=*/(short)0, acc, /*reuse_a=*/false, /*reuse_b=*/false);
    }

    // Column N=0 lives in lanes 0 (M=0..7 across VGPR 0..7) and 16 (M=8..15).
    if (lane == 0 || lane == 16) {
        const int moff = (lane == 0) ? 0 : 8;
#pragma unroll
        for (int r = 0; r < 8; ++r) {
            const int j = jbase + moff + r;
            const int o = a_out[b * kNAOut + j];
            out_values[b * kNAOut + j] = acc[r] + bias[o];
        }
    }
}

// Second tuple output: pass active_out_indices through.
__global__ void slide_copy_indices(const int* __restrict__ a_out,
                                   int* __restrict__ dst, int n)
{
    const int i = blockIdx.x * blockDim.x + threadIdx.x;
    if (i < n) dst[i] = a_out[i];
}

extern "C" void kernel_launch(void* const* d_in, const int* in_sizes, int n_in,
                              void* d_out, int out_size, void* d_ws, size_t ws_size,
                              hipStream_t stream)
{
    (void)in_sizes; (void)n_in; (void)out_size; (void)d_ws; (void)ws_size;

    const float* in_values = (const float*)d_in[0];
    const float* weight    = (const float*)d_in[1];
    const float* bias      = (const float*)d_in[2];
    const int*   a_in      = (const int*)d_in[3];
    const int*   a_out     = (const int*)d_in[4];

    float* out_values = (float*)d_out;
    int*   out_idx    = (int*)((float*)d_out + (size_t)kB * kNAOut);

    slide_gather_gemv_wmma<<<kB * kBlocksPerBatch, 256, 0, stream>>>(
        in_values, weight, bias, a_in, a_out, out_values);

    const int n = kB * kNAOut;
    slide_copy_indices<<<(n + 255) / 256, 256, 0, stream>>>(a_out, out_idx, n);
}